// GraphConvolution_2405181685968
// MI455X (gfx1250) — compile-verified
//
#include <hip/hip_runtime.h>

typedef __attribute__((ext_vector_type(2))) float v2f;
typedef __attribute__((ext_vector_type(8))) float v8f;

#define IN_DIM 256
#define OUT_DIM 128
#define LDS_STRIDE (IN_DIM + 4)   // 260 floats: breaks the stride-256 bank conflict
                                  // (260 % 64 == 4 -> 16 M-rows land on distinct bank groups)

// ---------------------------------------------------------------------------
// Kernel 1: hidden = fea @ W  using V_WMMA_F32_16X16X4_F32 (fp32 precision).
// Block = 256 threads (8 waves). Block owns 32 rows of fea (staged in LDS);
// wave w owns n-tile [16w, 16w+16) and computes TWO 16x16 C tiles (rows 0-15
// and 16-31 of the block tile), sharing each B fragment across both WMMAs.
// ---------------------------------------------------------------------------
__global__ __launch_bounds__(256) void gemm_wmma_kernel(
    const float* __restrict__ fea, const float* __restrict__ weight,
    float* __restrict__ hidden, int n_nodes) {
  __shared__ float4 ldsA4[32 * LDS_STRIDE / 4];  // 32 rows x 260 cols f32 (~33 KB)
  float* ldsA = (float*)ldsA4;

  const int tid = threadIdx.x;
  const int wave = tid >> 5;
  const int lane = tid & 31;
  const int m_base = blockIdx.x * 32;

  // Stage A tile into LDS with coalesced float4 loads (8 per thread).
  // Row r occupies float4 slots [r*65, r*65+64) (LDS_STRIDE/4 == 65).
  #pragma unroll
  for (int i = 0; i < 8; ++i) {
    int f = tid + i * 256;            // logical float4 index, 2048 total
    int r = f >> 6;                   // 64 float4 per row
    int kq = f & 63;
    int row = m_base + r;
    row = (row < n_nodes) ? row : (n_nodes - 1);
    ldsA4[r * (LDS_STRIDE / 4) + kq] =
        ((const float4*)(fea + (size_t)row * IN_DIM))[kq];
  }
  __syncthreads();

  const int mloc  = lane & 15;                 // A: M = lane % 16
  const int khalf = (lane >> 4) * 2;           // A/B: K sub-offset 0 or 2
  const int ncol  = wave * 16 + (lane & 15);   // B/C: N = n0 + lane%16
  const float* wcol = weight + ncol;

  v8f c0 = {};  // rows m_base + 0..15
  v8f c1 = {};  // rows m_base + 16..31

  #pragma unroll 4
  for (int k0 = 0; k0 < IN_DIM; k0 += 4) {
    v2f a0 = *(const v2f*)&ldsA[mloc * LDS_STRIDE + k0 + khalf];
    v2f a1 = *(const v2f*)&ldsA[(mloc + 16) * LDS_STRIDE + k0 + khalf];
    v2f b;
    b[0] = wcol[(size_t)(k0 + khalf) * OUT_DIM];
    b[1] = wcol[(size_t)(k0 + khalf + 1) * OUT_DIM];
    c0 = __builtin_amdgcn_wmma_f32_16x16x4_f32(false, a0, false, b,
                                               (short)0, c0, false, false);
    c1 = __builtin_amdgcn_wmma_f32_16x16x4_f32(false, a1, false, b,
                                               (short)0, c1, false, false);
  }

  // C/D layout: VGPR i -> row = i + 8*(lane>>4), col = lane&15 (per 16x16 tile)
  const int rhalf = (lane >> 4) * 8;
  float* base = hidden + (size_t)(m_base + rhalf) * OUT_DIM + ncol;
  if (m_base + 32 <= n_nodes) {
    // Fast path: one clause of 16 stores with immediate offsets.
    #pragma unroll
    for (int i = 0; i < 8; ++i) {
      base[(size_t)i * OUT_DIM] = c0[i];
      base[(size_t)(i + 16) * OUT_DIM] = c1[i];
    }
  } else {
    #pragma unroll
    for (int i = 0; i < 8; ++i) {
      if (m_base + rhalf + i < n_nodes)      base[(size_t)i * OUT_DIM] = c0[i];
      if (m_base + rhalf + i + 16 < n_nodes) base[(size_t)(i + 16) * OUT_DIM] = c1[i];
    }
  }
}

// ---------------------------------------------------------------------------
// Kernel 2: out[i][n] = bias[n]  (float4 vectorized init)
// ---------------------------------------------------------------------------
__global__ __launch_bounds__(256) void init_bias_kernel(
    const float* __restrict__ bias, float* __restrict__ out, long long total4) {
  long long i = (long long)blockIdx.x * 256 + threadIdx.x;
  if (i < total4) {
    ((float4*)out)[i] = ((const float4*)bias)[i & (OUT_DIM / 4 - 1)];
  }
}

// ---------------------------------------------------------------------------
// Kernel 3: per-edge scatter. One wave32 per edge: lane L gathers
// hidden[col][4L..4L+3] (coalesced 512B per wave, L2-resident), scales by
// edge_val, and does 4 native fp32 atomic adds into out[row] (coalesced).
// Edge metadata is wave-uniform: route the index through readfirstlane so the
// row/col/val loads become scalar (s_load via constant cache).
// ---------------------------------------------------------------------------
__global__ __launch_bounds__(256) void spmm_scatter_kernel(
    const int* __restrict__ erow, const int* __restrict__ ecol,
    const float* __restrict__ eval, const float* __restrict__ hidden,
    float* __restrict__ out, int n_edges) {
  int gwave = (int)(((long long)blockIdx.x * 256 + threadIdx.x) >> 5);
  int lane = threadIdx.x & 31;
  if (gwave >= n_edges) return;

  int e = __builtin_amdgcn_readfirstlane(gwave);  // wave-uniform edge id
  int r = erow[e];
  int c = ecol[e];
  float v = eval[e];

  float4 h = ((const float4*)(hidden + (size_t)c * OUT_DIM))[lane];
  float* o = out + (size_t)r * OUT_DIM + lane * 4;
  unsafeAtomicAdd(o + 0, v * h.x);   // global_atomic_add_f32, no CAS loop
  unsafeAtomicAdd(o + 1, v * h.y);
  unsafeAtomicAdd(o + 2, v * h.z);
  unsafeAtomicAdd(o + 3, v * h.w);
}

// ---------------------------------------------------------------------------
extern "C" void kernel_launch(void* const* d_in, const int* in_sizes, int n_in,
                              void* d_out, int out_size, void* d_ws, size_t ws_size,
                              hipStream_t stream) {
  const float* fea    = (const float*)d_in[0];
  const int*   erow   = (const int*)  d_in[1];
  const int*   ecol   = (const int*)  d_in[2];
  const float* eval   = (const float*)d_in[3];
  const float* weight = (const float*)d_in[4];
  const float* bias   = (const float*)d_in[5];
  float* out    = (float*)d_out;
  float* hidden = (float*)d_ws;   // n_nodes * 128 * 4 bytes = 51.2 MB scratch

  const int n_nodes = in_sizes[0] / IN_DIM;
  const int n_edges = in_sizes[1];

  // 1) hidden = fea @ W  (WMMA fp32)
  int gemm_blocks = (n_nodes + 31) / 32;
  gemm_wmma_kernel<<<gemm_blocks, 256, 0, stream>>>(fea, weight, hidden, n_nodes);

  // 2) out = broadcast(bias)
  long long total4 = (long long)n_nodes * OUT_DIM / 4;
  int ib = (int)((total4 + 255) / 256);
  init_bias_kernel<<<ib, 256, 0, stream>>>(bias, out, total4);

  // 3) out[row] += val * hidden[col]  (one wave per edge)
  long long nwaves = n_edges;
  int sb = (int)((nwaves + 7) / 8);
  spmm_scatter_kernel<<<sb, 256, 0, stream>>>(erow, ecol, eval, hidden, out, n_edges);
}